// Gridding_37873021616737
// MI455X (gfx1250) — compile-verified
//
#include <hip/hip_runtime.h>
#include <cstddef>
#include <cstdint>

// ---------------------------------------------------------------------------
// Gridding (trilinear scatter-add) for MI455X / gfx1250.
//   ptcloud [B,N,3] f32, scale s (=32) -> grid [B, (2s)^3] f32
// Atomic-bound kernel: 8 global_atomic_add_f32 per point into a 32MB output
// that lives in the 192MB L2. Point stream staged into LDS via the CDNA5
// async global->LDS path (ASYNCcnt), double buffered.
// ---------------------------------------------------------------------------

#define THREADS          256
#define TILE_PTS         1024
#define TILE_FLTS        (TILE_PTS * 3)           // 3072 floats = 12 KB
#define TILE_BYTES       (TILE_FLTS * 4)
#define TILES_PER_BLOCK  4
#define CHUNK_PTS        (TILE_PTS * TILES_PER_BLOCK)   // 4096 points per block
#define ASYNC_OPS_PER_TILE (TILE_BYTES / (THREADS * 16)) // 3 b128 ops per lane

#if defined(__HIP_DEVICE_COMPILE__)
#if __has_builtin(__builtin_amdgcn_global_load_async_to_lds_b128)
#define HAVE_ASYNC_LDS 1
#warning CDNA5 probe: __builtin_amdgcn_global_load_async_to_lds_b128 AVAILABLE
#else
#define HAVE_ASYNC_LDS 0
#warning CDNA5 probe: __builtin_amdgcn_global_load_async_to_lds_b128 MISSING - fallback copy path
#endif
#else
#define HAVE_ASYNC_LDS 0
#endif

#if defined(__HIP_DEVICE_COMPILE__) && __has_builtin(__builtin_amdgcn_s_wait_asynccnt)
#define WAIT_ASYNC(n) __builtin_amdgcn_s_wait_asynccnt(n)
#else
#define WAIT_ASYNC(n) asm volatile("s_wait_asynccnt " #n ::: "memory")
#endif

#define AS_GLOBAL __attribute__((address_space(1)))
#define AS_LOCAL  __attribute__((address_space(3)))

typedef int v4i __attribute__((ext_vector_type(4)));

// Stage one 12KB tile (1024 points * 3 floats) from global into LDS.
// Each of the 256 threads moves 3 x 16B chunks. With the async path each wave
// issues 3 ASYNCcnt-tracked copies; completion enforced by WAIT_ASYNC.
__device__ __forceinline__ void stage_tile(const float* __restrict__ gsrc,
                                           float* lbuf, int tid) {
#if HAVE_ASYNC_LDS
  char* g = (char*)gsrc;   // builtin takes non-const addrspace(1) v4i*
  char* l = (char*)lbuf;
#pragma unroll
  for (int r = 0; r < ASYNC_OPS_PER_TILE; ++r) {
    int off = (r * THREADS + tid) * 16;
    __builtin_amdgcn_global_load_async_to_lds_b128(
        (AS_GLOBAL v4i*)(g + off),
        (AS_LOCAL v4i*)(l + off),
        0, 0);
  }
#else
#pragma unroll
  for (int r = 0; r < ASYNC_OPS_PER_TILE; ++r) {
    int i4 = r * THREADS + tid;
    reinterpret_cast<float4*>(lbuf)[i4] =
        reinterpret_cast<const float4*>(gsrc)[i4];
  }
#endif
}

__global__ void __launch_bounds__(THREADS)
gridding_scatter(const float* __restrict__ pt, const int* __restrict__ scale_p,
                 float* __restrict__ out, int N) {
  const int s = *scale_p;            // uniform scalar load (s == 32 here)
  const float sf = (float)s;
  const int G  = 2 * s;              // 64
  const int G2 = G * G;              // 4096
  const size_t G3 = (size_t)G2 * G;  // 262144

  const int b   = blockIdx.y;
  const int tid = threadIdx.x;
  const float* gbase =
      pt + ((size_t)b * (size_t)N + (size_t)blockIdx.x * CHUNK_PTS) * 3;
  float* obase = out + (size_t)b * G3;

  __shared__ __align__(16) float smem[2][TILE_FLTS];

  // Prime the pipeline.
  stage_tile(gbase, smem[0], tid);

  for (int t = 0; t < TILES_PER_BLOCK; ++t) {
    if (t + 1 < TILES_PER_BLOCK) {
      // Kick off next tile, then wait until only its copies are outstanding.
      stage_tile(gbase + (size_t)(t + 1) * TILE_FLTS, smem[(t + 1) & 1], tid);
#if HAVE_ASYNC_LDS
      WAIT_ASYNC(ASYNC_OPS_PER_TILE);
#endif
    } else {
#if HAVE_ASYNC_LDS
      WAIT_ASYNC(0);
#endif
    }
    __syncthreads();  // LDS tile t visible to all waves in the block

    const float* lp = smem[t & 1];
#pragma unroll
    for (int k = 0; k < TILE_PTS / THREADS; ++k) {
      const int p = tid + k * THREADS;     // stride-256 -> conflict-light LDS
      const float px = lp[3 * p + 0] * sf;
      const float py = lp[3 * p + 1] * sf;
      const float pz = lp[3 * p + 2] * sf;

      // Reference drops points whose scaled coord-sum is exactly zero.
      if (px + py + pz == 0.0f) continue;

      const float fx = floorf(px), fy = floorf(py), fz = floorf(pz);
      int ix = (int)fx + s, iy = (int)fy + s, iz = (int)fz + s;
      ix = min(max(ix, 0), G - 2);
      iy = min(max(iy, 0), G - 2);
      iz = min(max(iz, 0), G - 2);

      const float ax = px - fx, ay = py - fy, az = pz - fz;  // upper weights
      const float bx = 1.0f - ax, by = 1.0f - ay, bz = 1.0f - az;

      float* dst = obase + ((size_t)ix * G + iy) * G + iz;
      const float bxby = bx * by, bxay = bx * ay;
      const float axby = ax * by, axay = ax * ay;
      atomicAdd(dst,              bxby * bz);
      atomicAdd(dst + 1,          bxby * az);
      atomicAdd(dst + G,          bxay * bz);
      atomicAdd(dst + G + 1,      bxay * az);
      atomicAdd(dst + G2,         axby * bz);
      atomicAdd(dst + G2 + 1,     axby * az);
      atomicAdd(dst + G2 + G,     axay * bz);
      atomicAdd(dst + G2 + G + 1, axay * az);
    }
    __syncthreads();  // done reading smem[t&1] before it is restaged (t+2)
  }
}

extern "C" void kernel_launch(void* const* d_in, const int* in_sizes, int n_in,
                              void* d_out, int out_size, void* d_ws, size_t ws_size,
                              hipStream_t stream) {
  (void)n_in; (void)d_ws; (void)ws_size;
  const float* pt      = (const float*)d_in[0];
  const int*   scale_p = (const int*)d_in[1];
  float*       out     = (float*)d_out;

  // Harness fixes scale=32 -> G=64, G^3=262144. Derive B, N from flat sizes.
  const int G3 = 64 * 64 * 64;
  const int B  = out_size / G3;                        // 32
  const long long npts = (long long)in_sizes[0] / 3;   // B*N
  const int N = (int)(npts / B);                       // 262144 (divisible by 4096)
  const int chunks = N / CHUNK_PTS;                    // 64 blocks per batch

  // Scatter target must start from zero (d_out is poisoned by the harness).
  (void)hipMemsetAsync(out, 0, (size_t)out_size * sizeof(float), stream);

  dim3 grid(chunks, B), block(THREADS);
  hipLaunchKernelGGL(gridding_scatter, grid, block, 0, stream,
                     pt, scale_p, out, N);
}